// Model_48335561949210
// MI455X (gfx1250) — compile-verified
//
#include <hip/hip_runtime.h>

// RGCN(4 layers) + GAT(1 head) + SumPool + dense head for gfx1250 (MI455X).
// Dense parts use V_WMMA_F32_16X16X4_F32 (full f32 precision); edge parts use
// L2-resident gather + global_atomic_add_f32 scatter.

#define N_NODES 50000
#define N_EDGES 1600000
#define DIM     64
#define NREL    4
#define NLAYERS 4
#define NGRAPHS 256
#define NEG_SLOPE 0.2f

typedef __attribute__((ext_vector_type(2))) float v2f;
typedef __attribute__((ext_vector_type(8))) float v8f;

// ---------------------------------------------------------------------------
// WMMA GEMM: OUT[row, col] = sum_d H[row,d] * W(d,col) (+ bias[col])
//   one wave (32 threads) per 16x16 output tile; K=64 via 16x V_WMMA_F32_16X16X4_F32
//   REL=false: W row-major [64 x 64], element W[d*64 + col]
//   REL=true : W rel-packed [R,64,64]; col = r*64+o -> W[(r*64+d)*64 + o]
// In both modes the per-lane B column reduces to a single base pointer plus
// compile-time offsets (k*64 floats), so the K-loop is branch-free with
// immediate-offset global loads.
// A layout (16x4 f32): lanes 0-15 = rows M, VGPR0/1 = K{0,1}; lanes 16-31 = K{2,3}
// B layout (4x16 f32): lanes = cols N, VGPR0/1 = K{0,1} (lo half) / K{2,3} (hi half)
// C/D layout: VGPR v holds M = v + 8*(lane>=16), N = lane&15
// ---------------------------------------------------------------------------
template <bool REL>
__global__ __launch_bounds__(32)
void gemm_wmma(const float* __restrict__ H, const float* __restrict__ W,
               const float* __restrict__ bias, float* __restrict__ OUT,
               int out_ld)
{
  const int lane = threadIdx.x;
  const int row0 = blockIdx.x * 16;
  const int col0 = blockIdx.y * 16;
  const int m    = lane & 15;
  const int hi   = lane >> 4;      // 0 or 1 (half-wave)
  const int koff = hi * 2;

  const float* __restrict__ hrow = H + (size_t)(row0 + m) * DIM + koff;
  const int c = col0 + m;

  // per-lane B column base (element index for K row = koff)
  const float* __restrict__ wcol;
  if (REL) wcol = W + (size_t)((c >> 6) * 64 + koff) * 64 + (c & 63);
  else     wcol = W + (size_t)koff * 64 + c;

  v8f acc = {};
#pragma unroll
  for (int s = 0; s < 16; ++s) {
    v2f a = *(const v2f*)(hrow + s * 4);   // h[row, 4s+koff], h[row, 4s+koff+1]
    v2f b;
    b.x = wcol[s * 256];                   // W(4s+koff,   c)
    b.y = wcol[s * 256 + 64];              // W(4s+koff+1, c)
    acc = __builtin_amdgcn_wmma_f32_16x16x4_f32(false, a, false, b,
                                                (short)0, acc, false, false);
  }

  const float bv = bias ? bias[c] : 0.0f;
#pragma unroll
  for (int v = 0; v < 8; ++v) {
    const int mrow = v + 8 * hi;
    OUT[(size_t)(row0 + mrow) * out_ld + c] = acc[v] + bv;
  }
}

// ---------------------------------------------------------------------------
// RGCN edge scatter: agg[dst] += hr[src, etype]; one wave per edge, 2 ch/lane.
// ---------------------------------------------------------------------------
__global__ __launch_bounds__(256)
void rgcn_edges(const float* __restrict__ hr, const int* __restrict__ src,
                const int* __restrict__ dst, const int* __restrict__ et,
                float* __restrict__ agg)
{
  const int gid  = blockIdx.x * blockDim.x + threadIdx.x;
  const int e    = gid >> 5;
  const int lane = gid & 31;
  if (e >= N_EDGES) return;
  const int s = src[e], d = dst[e], r = et[e];
  const v2f v = *(const v2f*)(hr + ((size_t)s * NREL + r) * DIM + 2 * lane);
  float* out = agg + (size_t)d * DIM + 2 * lane;
  atomicAdd(out,     v.x);
  atomicAdd(out + 1, v.y);
}

__global__ __launch_bounds__(256)
void relu_kernel(const float* __restrict__ in, float* __restrict__ out, int n)
{
  int i = blockIdx.x * blockDim.x + threadIdx.x;
  if (i < n) out[i] = fmaxf(in[i], 0.0f);
}

// ---------------------------------------------------------------------------
// GAT helpers
// ---------------------------------------------------------------------------
__device__ __forceinline__ float leaky(float x) { return x > 0.0f ? x : NEG_SLOPE * x; }

__device__ __forceinline__ void atomicMaxF(float* a, float v)
{
  if (v >= 0.0f) atomicMax((int*)a, __float_as_int(v));
  else           atomicMin((unsigned int*)a, __float_as_uint(v));
}

// el[n] = z[n].al ; er[n] = z[n].ar  (wave per node, shfl_xor reduce, wave32)
__global__ __launch_bounds__(256)
void gat_node_scores(const float* __restrict__ z, const float* __restrict__ al,
                     const float* __restrict__ ar, float* __restrict__ el,
                     float* __restrict__ er)
{
  const int gid  = blockIdx.x * blockDim.x + threadIdx.x;
  const int n    = gid >> 5;
  const int lane = gid & 31;
  if (n >= N_NODES) return;
  const float z0 = z[(size_t)n * DIM + lane];
  const float z1 = z[(size_t)n * DIM + 32 + lane];
  float pl = z0 * al[lane] + z1 * al[lane + 32];
  float pr = z0 * ar[lane] + z1 * ar[lane + 32];
#pragma unroll
  for (int off = 16; off > 0; off >>= 1) {
    pl += __shfl_xor(pl, off, 32);
    pr += __shfl_xor(pr, off, 32);
  }
  if (lane == 0) { el[n] = pl; er[n] = pr; }
}

__global__ __launch_bounds__(256)
void gat_init(float* __restrict__ hg, float* __restrict__ m,
              float* __restrict__ den, float* __restrict__ out,
              const float* __restrict__ bd)
{
  int i = blockIdx.x * blockDim.x + threadIdx.x;
  if (i < N_NODES * DIM) hg[i] = 0.0f;
  if (i < N_NODES) { m[i] = -__builtin_inff(); den[i] = 0.0f; }
  if (i < NGRAPHS) out[i] = bd[0];
}

__global__ __launch_bounds__(256)
void gat_edge_max(const int* __restrict__ src, const int* __restrict__ dst,
                  const float* __restrict__ el, const float* __restrict__ er,
                  float* __restrict__ m)
{
  int e = blockIdx.x * blockDim.x + threadIdx.x;
  if (e >= N_EDGES) return;
  const int d = dst[e];
  atomicMaxF(&m[d], leaky(el[src[e]] + er[d]));
}

__global__ __launch_bounds__(256)
void gat_edge_den(const int* __restrict__ src, const int* __restrict__ dst,
                  const float* __restrict__ el, const float* __restrict__ er,
                  const float* __restrict__ m, float* __restrict__ den)
{
  int e = blockIdx.x * blockDim.x + threadIdx.x;
  if (e >= N_EDGES) return;
  const int d = dst[e];
  const float v = leaky(el[src[e]] + er[d]);
  atomicAdd(&den[d], expf(v - m[d]));
}

// hg[dst] += alpha * z[src]; one wave per edge, 2 channels per lane
__global__ __launch_bounds__(256)
void gat_edge_agg(const int* __restrict__ src, const int* __restrict__ dst,
                  const float* __restrict__ el, const float* __restrict__ er,
                  const float* __restrict__ m, const float* __restrict__ den,
                  const float* __restrict__ z, float* __restrict__ hg)
{
  const int gid  = blockIdx.x * blockDim.x + threadIdx.x;
  const int e    = gid >> 5;
  const int lane = gid & 31;
  if (e >= N_EDGES) return;
  const int s = src[e], d = dst[e];
  const float alpha = expf(leaky(el[s] + er[d]) - m[d]) / fmaxf(den[d], 1e-9f);
  const v2f v = *(const v2f*)(z + (size_t)s * DIM + 2 * lane);
  float* out = hg + (size_t)d * DIM + 2 * lane;
  atomicAdd(out,     alpha * v.x);
  atomicAdd(out + 1, alpha * v.y);
}

// out[gids[n]] += hg[n] . Wd   (wave per node)
__global__ __launch_bounds__(256)
void pool_head(const float* __restrict__ hg, const float* __restrict__ Wd,
               const int* __restrict__ gids, float* __restrict__ out)
{
  const int gid  = blockIdx.x * blockDim.x + threadIdx.x;
  const int n    = gid >> 5;
  const int lane = gid & 31;
  if (n >= N_NODES) return;
  float v = hg[(size_t)n * DIM + lane] * Wd[lane]
          + hg[(size_t)n * DIM + 32 + lane] * Wd[lane + 32];
#pragma unroll
  for (int off = 16; off > 0; off >>= 1) v += __shfl_xor(v, off, 32);
  if (lane == 0) atomicAdd(&out[gids[n]], v);
}

// ---------------------------------------------------------------------------
extern "C" void kernel_launch(void* const* d_in, const int* in_sizes, int n_in,
                              void* d_out, int out_size, void* d_ws, size_t ws_size,
                              hipStream_t stream)
{
  const float* h0    = (const float*)d_in[0];
  const float* Wrel  = (const float*)d_in[1];
  const float* Wloop = (const float*)d_in[2];
  const float* brel  = (const float*)d_in[3];
  const float* Wg    = (const float*)d_in[4];
  const float* al    = (const float*)d_in[5];
  const float* ar    = (const float*)d_in[6];
  const float* Wd    = (const float*)d_in[7];
  const float* bd    = (const float*)d_in[8];
  const int*   src   = (const int*)d_in[9];
  const int*   dst   = (const int*)d_in[10];
  const int*   et    = (const int*)d_in[11];
  const int*   gids  = (const int*)d_in[12];
  float*       out   = (float*)d_out;

  float* ws    = (float*)d_ws;
  float* h_cur = ws;                                  // N*D
  float* agg   = h_cur + (size_t)N_NODES * DIM;       // N*D  (reused as z)
  float* hr    = agg   + (size_t)N_NODES * DIM;       // N*R*D (reused as hg)
  float* elb   = hr    + (size_t)N_NODES * NREL * DIM;
  float* erb   = elb + N_NODES;
  float* mbuf  = erb + N_NODES;
  float* den   = mbuf + N_NODES;

  const dim3 wave(32);
  const int rowTiles        = N_NODES / 16;                 // 3125 (exact)
  const int edgeWaveBlocks  = (N_EDGES * 32) / 256;         // wave-per-edge kernels
  const int edgeThrBlocks   = (N_EDGES + 255) / 256;        // thread-per-edge kernels
  const int nodeWaveBlocks  = (N_NODES * 32 + 255) / 256;   // wave-per-node kernels
  const int nd              = N_NODES * DIM;

  const float* hin = h0;
  for (int l = 0; l < NLAYERS; ++l) {
    // hr[n, r*64+o] = sum_d h[n,d] * Wrel[l,r,d,o]   (single [N,64]x[64,256] GEMM)
    gemm_wmma<true><<<dim3(rowTiles, (NREL * DIM) / 16), wave, 0, stream>>>(
        hin, Wrel + (size_t)l * NREL * DIM * DIM, nullptr, hr, NREL * DIM);
    // agg = h @ Wloop[l] + brel[l]   (scatter accumulates on top of this)
    gemm_wmma<false><<<dim3(rowTiles, DIM / 16), wave, 0, stream>>>(
        hin, Wloop + (size_t)l * DIM * DIM, brel + (size_t)l * DIM, agg, DIM);
    rgcn_edges<<<edgeWaveBlocks, 256, 0, stream>>>(hr, src, dst, et, agg);
    relu_kernel<<<(nd + 255) / 256, 256, 0, stream>>>(agg, h_cur, nd);
    hin = h_cur;
  }

  float* z  = agg;  // agg free after final relu
  float* hg = hr;   // reuse first N*D of hr

  gemm_wmma<false><<<dim3(rowTiles, DIM / 16), wave, 0, stream>>>(h_cur, Wg, nullptr, z, DIM);
  gat_node_scores<<<nodeWaveBlocks, 256, 0, stream>>>(z, al, ar, elb, erb);
  gat_init<<<(nd + 255) / 256, 256, 0, stream>>>(hg, mbuf, den, out, bd);
  gat_edge_max<<<edgeThrBlocks, 256, 0, stream>>>(src, dst, elb, erb, mbuf);
  gat_edge_den<<<edgeThrBlocks, 256, 0, stream>>>(src, dst, elb, erb, mbuf, den);
  gat_edge_agg<<<edgeWaveBlocks, 256, 0, stream>>>(src, dst, elb, erb, mbuf, den, z, hg);
  pool_head<<<nodeWaveBlocks, 256, 0, stream>>>(hg, Wd, gids, out);

  (void)in_sizes; (void)n_in; (void)out_size; (void)ws_size;
}